// Articulation_86809878987287
// MI455X (gfx1250) — compile-verified
//
#include <hip/hip_runtime.h>

typedef __attribute__((ext_vector_type(2))) float v2f;
typedef __attribute__((ext_vector_type(8))) float v8f;

#define LRELU_SLOPE 0.01f

// =====================================================================
// Generic fp32 WMMA GEMM:  H = act(A[32 x Kdim] @ W[Kdim x Ncols] + bias)
// One wave per 16x16 output tile.  V_WMMA_F32_16X16X4_F32 layouts:
//   A frag (16x4):  lanes 0-15 -> rows, vgpr0=K0 / vgpr1=K1 ; lanes 16-31 K2/K3
//   B frag (4x16):  mirrored (N across lanes, K split across halves/vgprs)
//   C/D (16x16):    lane<16 -> M=v, N=lane ; lane>=16 -> M=v+8, N=lane-16
// Out-of-range columns (layer 3, N=40): clamp the load address (always
// valid memory) and simply never store those lanes -- no EXEC masking,
// no zeroing, keeps the inner loop branch-free.
// =====================================================================
__global__ void __launch_bounds__(128)
mlp_wmma_kernel(const float* __restrict__ A, const float* __restrict__ W,
                const float* __restrict__ bias, float* __restrict__ H,
                int Kdim, int Ncols, int nTilesN, int nTiles, int applyLrelu)
{
    const int wave = threadIdx.x >> 5;
    const int lane = threadIdx.x & 31;
    const int tile = blockIdx.x * (blockDim.x >> 5) + wave;
    if (tile >= nTiles) return;                  // wave-uniform: EXEC stays all-ones
    const int mt   = tile / nTilesN;
    const int nt   = tile - mt * nTilesN;
    const int half = lane >> 4;
    const int l    = lane & 15;
    const int row  = mt * 16 + l;
    const int col  = nt * 16 + l;
    const int colc = (col < Ncols) ? col : (Ncols - 1);   // clamped, always valid

    // branch-free pointer-bump operand streams
    const float* ap  = A + (size_t)row * Kdim + half * 2;           // A[row][kb], A[row][kb+1]
    const float* wp0 = W + (size_t)(half * 2) * Ncols + colc;       // W[kb  ][colc]
    const float* wp1 = wp0 + Ncols;                                 // W[kb+1][colc]
    const int    wstep = 4 * Ncols;

    v8f acc = {};
    for (int kk = 0; kk < Kdim; kk += 4) {
        v2f a = *(const v2f*)ap;                 // 8B-aligned global_load_b64
        v2f b;
        b.x = *wp0;
        b.y = *wp1;
        acc = __builtin_amdgcn_wmma_f32_16x16x4_f32(false, a, false, b,
                                                    (short)0, acc, false, false);
        ap  += 4;
        wp0 += wstep;
        wp1 += wstep;
    }
    if (col >= Ncols) return;                    // discard clamped garbage lanes
    const float bv    = bias[col];
    const int   rbase = mt * 16 + half * 8;
#pragma unroll
    for (int v = 0; v < 8; ++v) {
        float d = acc[v] + bv;
        if (applyLrelu) d = (d > 0.f) ? d : LRELU_SLOPE * d;
        H[(size_t)(rbase + v) * Ncols + col] = d;
    }
}

// =====================================================================
// Pose kernel: one thread per batch n (N=32).
// vec(32x40) -> axis-angle -> R -> kinematic chain -> T_all(32 x 96):
//   T_all[p*4+j][n*3+i] = Rg[n,p,i,j] (j<3) | tg[n,p,i] (j==3)
// Also zeros the t_net output region.
// =====================================================================
__global__ void pose_kernel(const float* __restrict__ vec,
                            const float* __restrict__ rot_center,
                            const float* __restrict__ axis,
                            float* __restrict__ Tall,
                            float* __restrict__ tnet)
{
    const int n = threadIdx.x;
    if (n >= 32) return;
    const int parent[8] = {-1, 0, 0, 1, 1, 2, 2, 3};
    float Rg[8][9];
    float tg[8][3];
#pragma unroll
    for (int p = 0; p < 8; ++p) {
        const float vx = vec[n * 40 + p * 5 + 0];
        const float vy = vec[n * 40 + p * 5 + 1];
        const float vz = vec[n * 40 + p * 5 + 2];
        const float nr = fmaxf(sqrtf(vx * vx + vy * vy + vz * vz), 1e-12f);
        const float angle = atan2f(vy / nr, vx / nr);

        float ax = axis[p * 3 + 0], ay = axis[p * 3 + 1], az = axis[p * 3 + 2];
        const float an = fmaxf(sqrtf(ax * ax + ay * ay + az * az), 1e-12f);
        const float lx = angle * ax / an;
        const float ly = angle * ay / an;
        const float lz = angle * az / an;

        const float nrms  = lx * lx + ly * ly + lz * lz;
        const float theta = sqrtf(fmaxf(nrms, 1e-4f));   // jnp.clip(nrms, eps)
        const float s = sinf(theta), c = cosf(theta);
        const float f1 = s / theta;
        const float f2 = (1.f - c) / (theta * theta);

        const float Km[9] = {0.f, -lz, ly,  lz, 0.f, -lx,  -ly, lx, 0.f};
        float K2[9];
#pragma unroll
        for (int i = 0; i < 3; ++i)
#pragma unroll
            for (int j = 0; j < 3; ++j) {
                float sum = 0.f;
#pragma unroll
                for (int t = 0; t < 3; ++t) sum += Km[i * 3 + t] * Km[t * 3 + j];
                K2[i * 3 + j] = sum;
            }
        float R[9];
#pragma unroll
        for (int e = 0; e < 9; ++e) R[e] = f1 * Km[e] + f2 * K2[e];
        R[0] += 1.f; R[4] += 1.f; R[8] += 1.f;

        const float rc[3] = {rot_center[p * 3], rot_center[p * 3 + 1], rot_center[p * 3 + 2]};
        float tp[3];
#pragma unroll
        for (int i = 0; i < 3; ++i)
            tp[i] = rc[i] - (R[i * 3] * rc[0] + R[i * 3 + 1] * rc[1] + R[i * 3 + 2] * rc[2]);

        const int pa = parent[p];                 // pa < p always -> ascending order OK
        if (pa < 0) {
#pragma unroll
            for (int e = 0; e < 9; ++e) Rg[p][e] = R[e];
#pragma unroll
            for (int i = 0; i < 3; ++i) tg[p][i] = tp[i];
        } else {
#pragma unroll
            for (int i = 0; i < 3; ++i)
#pragma unroll
                for (int j = 0; j < 3; ++j) {
                    float sum = 0.f;
#pragma unroll
                    for (int t = 0; t < 3; ++t) sum += Rg[pa][i * 3 + t] * R[t * 3 + j];
                    Rg[p][i * 3 + j] = sum;
                }
#pragma unroll
            for (int i = 0; i < 3; ++i)
                tg[p][i] = Rg[pa][i * 3 + 0] * tp[0] + Rg[pa][i * 3 + 1] * tp[1] +
                           Rg[pa][i * 3 + 2] * tp[2] + tg[pa][i];
        }
    }
#pragma unroll
    for (int p = 0; p < 8; ++p)
#pragma unroll
        for (int i = 0; i < 3; ++i) {
#pragma unroll
            for (int j = 0; j < 3; ++j)
                Tall[(p * 4 + j) * 96 + n * 3 + i] = Rg[p][i * 3 + j];
            Tall[(p * 4 + 3) * 96 + n * 3 + i] = tg[p][i];
        }
#pragma unroll
    for (int e = 0; e < 24; ++e) tnet[n * 24 + e] = 0.f;   // t_net = zeros
}

// =====================================================================
// Skinning: arti = Vaug(64000x32) @ T_all(32x96) via fp32 WMMA.
// Block = 4 waves = 64 vertices. A-fragments built in registers from
// alpha/verts (each K-chunk of 4 == one joint: {a*vx, a*vy | a*vz, a}).
// Results staged in LDS -> fully coalesced 768B output segments per n.
// Memory-bound: ~25 MB traffic => ~1.1us at 23.3 TB/s; WMMA keeps the
// ~220M FMAs off the VALU critical path.
// =====================================================================
__global__ void __launch_bounds__(128)
skin_kernel(const float* __restrict__ verts, const float* __restrict__ alpha,
            const float* __restrict__ Tall, float* __restrict__ out, int Kverts)
{
    __shared__ float Tlds[32 * 96];      // 12 KB: shared B operand
    __shared__ float stage[96 * 65];     // ~25 KB: [col][k_local], padded stride 65
    const int tid = threadIdx.x;
    for (int idx = tid; idx < 32 * 96; idx += 128) Tlds[idx] = Tall[idx];
    __syncthreads();

    const int wave = tid >> 5;
    const int lane = tid & 31;
    const int half = lane >> 4;
    const int l    = lane & 15;
    const int row  = blockIdx.x * 64 + wave * 16 + l;   // 64000 = 1000*64: no guard

    // speculative prefetch of the next block's per-vertex data (L2 resident)
    __builtin_prefetch(alpha + (size_t)(row + 64) * 8, 0, 0);
    __builtin_prefetch(verts + (size_t)(row + 64) * 3, 0, 0);

    // per-vertex data (duplicated across lane halves)
    const float4 a01 = *(const float4*)(alpha + (size_t)row * 8);
    const float4 a23 = *(const float4*)(alpha + (size_t)row * 8 + 4);
    float al[8] = {a01.x, a01.y, a01.z, a01.w, a23.x, a23.y, a23.z, a23.w};
    const float vx = verts[(size_t)row * 3 + 0];
    const float vy = verts[(size_t)row * 3 + 1];
    const float vz = verts[(size_t)row * 3 + 2];
    const float e0 = half ? vz : vx;     // A element 0 of each K-chunk for this half
    const float e1m = half ? 1.f : vy;   // A element 1 multiplier

    const int col = l;                    // within-tile column
#pragma unroll
    for (int t = 0; t < 6; ++t) {
        v8f acc = {};
        const int c0 = t * 16 + col;
#pragma unroll
        for (int c = 0; c < 8; ++c) {     // K-chunk c == joint p=c
            const float alc = al[c];
            v2f a, b;
            a.x = alc * e0;
            a.y = alc * e1m;
            const int kk = 4 * c + half * 2;
            b.x = Tlds[kk * 96 + c0];
            b.y = Tlds[(kk + 1) * 96 + c0];
            acc = __builtin_amdgcn_wmma_f32_16x16x4_f32(false, a, false, b,
                                                        (short)0, acc, false, false);
        }
        const int rl = wave * 16 + half * 8;
#pragma unroll
        for (int v = 0; v < 8; ++v)
            stage[c0 * 65 + rl + v] = acc[v];
    }
    __syncthreads();

    // Coalesced output: per n, a contiguous 64*3 = 192-float (768B) segment.
    const long kb3 = (long)blockIdx.x * 64 * 3;
    for (int n = 0; n < 32; ++n) {
        float* dst = out + (long)n * Kverts * 3 + kb3;
        for (int f = tid; f < 192; f += 128) {
            const int kl = f / 3;
            const int i  = f - kl * 3;
            dst[f] = stage[(n * 3 + i) * 65 + kl];
        }
    }
}

// =====================================================================
extern "C" void kernel_launch(void* const* d_in, const int* in_sizes, int n_in,
                              void* d_out, int out_size, void* d_ws, size_t ws_size,
                              hipStream_t stream)
{
    (void)in_sizes; (void)n_in; (void)out_size; (void)ws_size;
    const float* x     = (const float*)d_in[0];
    const float* W1    = (const float*)d_in[1];
    const float* b1    = (const float*)d_in[2];
    const float* W2    = (const float*)d_in[3];
    const float* b2    = (const float*)d_in[4];
    const float* W3    = (const float*)d_in[5];
    const float* b3    = (const float*)d_in[6];
    const float* verts = (const float*)d_in[7];
    const float* rotc  = (const float*)d_in[8];
    const float* alpha = (const float*)d_in[9];
    const float* axis  = (const float*)d_in[10];
    float* out = (float*)d_out;

    const int N = 32, F = 512, KV = 64000;
    float* ws   = (float*)d_ws;
    float* h1   = ws;                // 32*512
    float* h2   = h1 + N * F;        // 32*512
    float* vec  = h2 + N * F;        // 32*40
    float* Tall = vec + N * 40;      // 32*96

    // layer 1: h1 = lrelu(x@W1+b1)    (64 tiles -> 16 blocks x 4 waves)
    mlp_wmma_kernel<<<16, 128, 0, stream>>>(x,  W1, b1, h1, F, F, F / 16, 2 * (F / 16), 1);
    // layer 2: h2 = lrelu(h1@W2+b2)
    mlp_wmma_kernel<<<16, 128, 0, stream>>>(h1, W2, b2, h2, F, F, F / 16, 2 * (F / 16), 1);
    // layer 3: vec = h2@W3+b3  (Ncols=40 -> 3 N-tiles, 6 tiles total)
    mlp_wmma_kernel<<<2, 128, 0, stream>>>(h2, W3, b3, vec, F, 40, 3, 6, 0);

    // pose -> T_all, and zero t_net tail of the output
    float* tnet = out + (size_t)N * KV * 3;
    pose_kernel<<<1, 32, 0, stream>>>(vec, rotc, axis, Tall, tnet);

    // skinning GEMM
    skin_kernel<<<KV / 64, 128, 0, stream>>>(verts, alpha, Tall, out, KV);
}